// PositionalEncodingQuantum_65481071410800
// MI455X (gfx1250) — compile-verified
//
#include <hip/hip_runtime.h>
#include <hip/hip_bf16.h>

// out[t, e] = cos(t) * sum_q W[e,q] + b[e], computed as a true f32 GEMM
// [T x 8] @ [8 x 1024] using V_WMMA_F32_16X16X4_F32 (K=8 -> two WMMAs per
// 16x16 tile). Kernel is store-bandwidth bound (256 MB out @ 23.3 TB/s
// ~ 11.5 us floor); W (32KB) and b (4KB) stay resident in cache.

typedef __attribute__((ext_vector_type(2))) float v2f;
typedef __attribute__((ext_vector_type(8))) float v8f;

#define EMBED_DIM 1024
#define N_QUBITS 8
#define T_TOTAL 65536
#define MTILES_PER_BLOCK 8   // grid.y = 4096 / 8 = 512
#define NTILE_GROUPS 8       // grid.x = 64 n-tiles / 8 waves

__global__ __launch_bounds__(256) void
peq_wmma_kernel(const float* __restrict__ W,   // [1024, 8] row-major
                const float* __restrict__ bias, // [1024]
                float* __restrict__ out)        // [65536, 1024]
{
    const int lane = threadIdx.x & 31;
    const int wave = threadIdx.x >> 5;          // 0..7
    const int hi   = lane >> 4;                 // 0: lanes 0-15, 1: lanes 16-31
    const int nIdx = lane & 15;                 // column within tile == A row M

    // This wave's N tile (fixed for whole kernel): reuse B + bias across M loop
    const int n = (blockIdx.x * NTILE_GROUPS + wave) * 16 + nIdx;

    // B fragment: 4x16 f32, VGPR v holds row K = v + 2*hi, col N = lane&15.
    // B[k][n] = W[n][k]  (einsum 'btq,eq->bte').
    const float* wrow = W + n * N_QUBITS;
    const int kbase = 2 * hi;
    v2f b_lo, b_hi;
    b_lo.x = wrow[kbase + 0];
    b_lo.y = wrow[kbase + 1];
    b_hi.x = wrow[kbase + 4];
    b_hi.y = wrow[kbase + 5];

    // Bias depends only on the column -> same value in all 8 C registers.
    const float bval = bias[n];

    const int mtile0 = blockIdx.y * MTILES_PER_BLOCK;

    #pragma unroll
    for (int i = 0; i < MTILES_PER_BLOCK; ++i) {
        const int t0 = (mtile0 + i) * 16;

        // A fragment: 16x4 f32, lane holds row M = lane&15, VGPR v holds
        // K = v + 2*hi. pe[t][q] = cos(t) for every q -> both slots equal.
        const float ct = cosf((float)(t0 + nIdx));
        v2f a;
        a.x = ct;
        a.y = ct;

        // Seed accumulator with the bias so D = A*B + bias directly.
        v8f c;
        #pragma unroll
        for (int v = 0; v < 8; ++v) c[v] = bval;

        // K = 0..3 then K = 4..7 (f32 WMMA, exact vs reference math)
        c = __builtin_amdgcn_wmma_f32_16x16x4_f32(
                false, a, false, b_lo, (short)0, c, false, false);
        c = __builtin_amdgcn_wmma_f32_16x16x4_f32(
                false, a, false, b_hi, (short)0, c, false, false);

        // D layout: VGPR v, lanes 0-15 -> row t0+v, lanes 16-31 -> row t0+v+8,
        // col = n. Each register store = two contiguous 64B row segments.
        float* orow = out + (size_t)(t0 + hi * 8) * EMBED_DIM + n;
        #pragma unroll
        for (int v = 0; v < 8; ++v) {
            orow[(size_t)v * EMBED_DIM] = c[v];
        }
    }
}

extern "C" void kernel_launch(void* const* d_in, const int* in_sizes, int n_in,
                              void* d_out, int out_size, void* d_ws, size_t ws_size,
                              hipStream_t stream) {
    // d_in[0] = x  (unused: only its shape matters in the reference)
    const float* W = (const float*)d_in[1];   // [1024, 8]
    const float* b = (const float*)d_in[2];   // [1024]
    float* out = (float*)d_out;               // [1, 65536, 1024]

    dim3 grid(NTILE_GROUPS, T_TOTAL / 16 / MTILES_PER_BLOCK); // (8, 512)
    dim3 block(256);                                          // 8 waves
    peq_wmma_kernel<<<grid, block, 0, stream>>>(W, b, out);
}